// Mask2FormerLoss_43112881718014
// MI455X (gfx1250) — compile-verified
//
#include <hip/hip_runtime.h>
#include <hip/hip_bf16.h>
#include <math.h>

#define BB  4
#define QQ  100
#define CC  3
#define NN  10
#define HWD 65536

#define LOG2E_F 1.44269504088896f
#define LN2_F   0.69314718055995f

typedef __attribute__((ext_vector_type(2))) float v2f;
typedef __attribute__((ext_vector_type(4))) float v4f;
typedef __attribute__((ext_vector_type(8))) float v8f;

// log(1 + exp(-a)) for a >= 0, via raw v_exp_f32/v_log_f32.
// 1 + exp2(t<=0) is in (1,2]: no denorm/inf guards needed.
__device__ __forceinline__ float l1pexp_na(float a) {
    return LN2_F * __builtin_amdgcn_logf(1.f + __builtin_amdgcn_exp2f(a * -LOG2E_F));
}
// softplus(x) = log(1+exp(x)) = max(x,0) + log(1+exp(-|x|))
__device__ __forceinline__ float softplus_fast(float x) {
    return fmaxf(x, 0.f) + l1pexp_na(fabsf(x));
}
// sigmoid via v_exp_f32 + v_rcp_f32
__device__ __forceinline__ float sigmoid_fast(float x) {
    return __builtin_amdgcn_rcpf(1.f + __builtin_amdgcn_exp2f(x * -LOG2E_F));
}

// ---------------------------------------------------------------------------
// Kernel A: cost[b,q,n] = -(pm.t + sum(log_sigmoid(-pm)))/HW - probs[b,q,lab[n]]
// V_WMMA_F32_16X16X4_F32 for the pm . t matmul (full f32 precision).
// grid = (7 q-tiles, 4 batches); 256 threads = 8 waves, K split across waves.
// ---------------------------------------------------------------------------
__global__ __launch_bounds__(256) void cost_kernel(
    const float* __restrict__ logits,   // (B,Q,C)
    const float* __restrict__ pm,       // (B,Q,HW)  mask logits
    const int*   __restrict__ labels,   // (B,N)
    const float* __restrict__ tmk,      // (B,N,HW)  raw target masks
    float*       __restrict__ cost)     // (B,Q,N)
{
    const int b    = blockIdx.y;
    const int q0   = blockIdx.x << 4;          // 16-row q tile
    const int tid  = threadIdx.x;
    const int w    = tid >> 5;                 // wave id 0..7
    const int lane = tid & 31;
    const int rowL = lane & 15;                // A-matrix row / D-matrix col
    const int half = lane >> 4;                // K-half select for A/B frags
    const int q    = q0 + rowL;
    const bool qok = (q < QQ);
    const bool nok = (rowL < NN);

    // Clamp pointers for out-of-range rows/cols: their WMMA outputs are never
    // stored, and their log-sigmoid contribution is masked out below.
    const float* __restrict__ pmRow = pm  + (size_t)(b * QQ + (qok ? q    : 0)) * HWD;
    const float* __restrict__ tmRow = tmk + (size_t)(b * NN + (nok ? rowL : 0)) * HWD;

    v8f   acc   = {0.f, 0.f, 0.f, 0.f, 0.f, 0.f, 0.f, 0.f};
    float accnp = 0.f;

    const int kbeg = w * (HWD / 8);
    const int kend = kbeg + (HWD / 8);
    const int base = half * 2;

#pragma unroll 8
    for (int kk = kbeg; kk < kend; kk += 4) {
        const int ko = kk + base;                       // 8B-aligned
        v2f a  = *(const v2f*)(pmRow + ko);             // A frag: pm[q, ko..ko+1]
        v2f tv = *(const v2f*)(tmRow + ko);
        v2f bf;
        bf.x = (tv.x > 0.f) ? 1.f : 0.f;                // B frag: binarized target
        bf.y = (tv.y > 0.f) ? 1.f : 0.f;
        // D = A(16x4) x B(4x16) + C, f32 throughout
        acc = __builtin_amdgcn_wmma_f32_16x16x4_f32(
            false, a, false, bf, (short)0, acc, false, false);
        // row-sum of log_sigmoid(-pm) = -softplus(pm); each element touched once
        accnp -= softplus_fast(a.x) + softplus_fast(a.y);
    }

    __shared__ float ldsd[8 * 256];     // per-wave partial D tiles
    __shared__ float ldsn[8 * 32];      // per-wave partial log-sigmoid sums
    __shared__ float lds_sum[16];       // per-row total log-sigmoid sum
    __shared__ float lds_prob[16][4];   // per-row softmax probs

#pragma unroll
    for (int v = 0; v < 8; ++v) ldsd[w * 256 + lane * 8 + v] = acc[v];
    ldsn[w * 32 + lane] = qok ? accnp : 0.f;
    __syncthreads();

    float tot[8];
    if (tid < 32) {
#pragma unroll
        for (int v = 0; v < 8; ++v) {
            float s = 0.f;
            for (int w2 = 0; w2 < 8; ++w2) s += ldsd[w2 * 256 + lane * 8 + v];
            tot[v] = s;
        }
        float npv = 0.f;
        for (int w2 = 0; w2 < 8; ++w2) npv += ldsn[w2 * 32 + lane];
        ldsn[lane] = npv;               // reuse slot 0..31 for combined np
    }
    __syncthreads();

    if (tid < 16) {
        lds_sum[tid] = ldsn[tid] + ldsn[tid + 16];      // join K-halves
        float p0 = 0.f, p1 = 0.f, p2 = 0.f;
        if (q0 + tid < QQ) {
            const float* lg = logits + (size_t)(b * QQ + q0 + tid) * CC;
            float l0 = lg[0], l1 = lg[1], l2 = lg[2];
            float mx = fmaxf(l0, fmaxf(l1, l2));
            float e0 = __builtin_amdgcn_exp2f((l0 - mx) * LOG2E_F);
            float e1 = __builtin_amdgcn_exp2f((l1 - mx) * LOG2E_F);
            float e2 = __builtin_amdgcn_exp2f((l2 - mx) * LOG2E_F);
            float inv = __builtin_amdgcn_rcpf(e0 + e1 + e2);
            p0 = e0 * inv; p1 = e1 * inv; p2 = e2 * inv;
        }
        lds_prob[tid][0] = p0; lds_prob[tid][1] = p1; lds_prob[tid][2] = p2;
    }
    __syncthreads();

    if (tid < 32) {
        const float invHW = 1.f / (float)HWD;
#pragma unroll
        for (int v = 0; v < 8; ++v) {
            // C/D layout: VGPR v, lanes 0-15 -> M=v, lanes 16-31 -> M=v+8
            int row = v + half * 8;
            int qq2 = q0 + row;
            int col = rowL;
            if (qq2 < QQ && col < NN) {
                int lab = labels[b * NN + col];
                float cm = -(lds_sum[row] + tot[v]) * invHW;
                cost[(size_t)(b * QQ + qq2) * NN + col] = cm - lds_prob[row][lab];
            }
        }
    }
}

// ---------------------------------------------------------------------------
// Kernel B: Jonker-Volgenant assignment on cost^T (10 targets x 100 queries),
// one serial thread per batch, double precision (matches numpy reference).
// ---------------------------------------------------------------------------
__global__ void hungarian_kernel(const float* __restrict__ cost,
                                 int* __restrict__ qof, int* __restrict__ tof)
{
    const int b = blockIdx.x;
    if (threadIdx.x != 0) return;
    const int n = NN, m = QQ;
    double u[NN + 1], v[QQ + 1], minv[QQ + 1];
    int p[QQ + 1], way[QQ + 1];
    bool used[QQ + 1];
    for (int i = 0; i <= n; ++i) u[i] = 0.0;
    for (int j = 0; j <= m; ++j) { v[j] = 0.0; p[j] = 0; way[j] = 0; }
    const float* cb = cost + (size_t)b * QQ * NN;   // costT(i,j) = cb[j*NN + i]

    for (int i = 1; i <= n; ++i) {
        p[0] = i;
        int j0 = 0;
        for (int j = 0; j <= m; ++j) { minv[j] = 1e18; used[j] = false; }
        while (true) {
            used[j0] = true;
            int i0 = p[j0];
            double delta = 1e18;
            int j1 = 0;
            for (int j = 1; j <= m; ++j) {
                if (!used[j]) {
                    double cur = (double)cb[(j - 1) * NN + (i0 - 1)] - u[i0] - v[j];
                    if (cur < minv[j]) { minv[j] = cur; way[j] = j0; }
                    if (minv[j] < delta) { delta = minv[j]; j1 = j; }
                }
            }
            for (int j = 0; j <= m; ++j) {
                if (used[j]) { u[p[j]] += delta; v[j] -= delta; }
                else         { minv[j] -= delta; }
            }
            j0 = j1;
            if (p[j0] == 0) break;
        }
        while (j0) {
            int jp = way[j0];
            p[j0] = p[jp];
            j0 = jp;
        }
    }
    int cnt = 0;
    for (int j = 1; j <= m; ++j) {
        if (p[j]) {
            qof[b * NN + cnt] = j - 1;      // matched query
            tof[b * NN + cnt] = p[j] - 1;   // matched target
            ++cnt;
        }
    }
}

// ---------------------------------------------------------------------------
// Kernel C: per matched pair, reduce sigmoid/dice/BCE sums over HW, and CE.
// grid = B*N blocks of 256 threads; float4 (b128) loads.
// ---------------------------------------------------------------------------
__global__ __launch_bounds__(256) void pair_kernel(
    const float* __restrict__ logits,
    const float* __restrict__ pm,
    const int*   __restrict__ labels,
    const float* __restrict__ tmk,
    const int*   __restrict__ qof, const int* __restrict__ tof,
    float* __restrict__ sums,       // (B*N,4): inter, psum, tsum, bce
    float* __restrict__ cevals)     // (B*N,)
{
    const int pairIdx = blockIdx.x;
    const int b = pairIdx / NN;
    const int q = qof[pairIdx];
    const int t = tof[pairIdx];
    const float* x  = pm  + (size_t)(b * QQ + q) * HWD;
    const float* mr = tmk + (size_t)(b * NN + t) * HWD;

    float inter = 0.f, ps = 0.f, ts = 0.f, bce = 0.f;
    for (int h = threadIdx.x * 4; h < HWD; h += 256 * 4) {
        v4f xv = *(const v4f*)(x + h);
        v4f mv = *(const v4f*)(mr + h);
#pragma unroll
        for (int e = 0; e < 4; ++e) {
            float xi = xv[e], mi = mv[e];
            float p = sigmoid_fast(xi);
            inter += p * mi;
            ps    += p;
            ts    += mi;
            // bce term: tb ? softplus(-x) : softplus(x); shared log1pexp part
            float L = l1pexp_na(fabsf(xi));
            bce += ((mi > 0.f) ? fmaxf(-xi, 0.f) : fmaxf(xi, 0.f)) + L;
        }
    }

    __shared__ float red[4][256];
    red[0][threadIdx.x] = inter; red[1][threadIdx.x] = ps;
    red[2][threadIdx.x] = ts;    red[3][threadIdx.x] = bce;
    __syncthreads();
    for (int s = 128; s > 0; s >>= 1) {
        if (threadIdx.x < s) {
            red[0][threadIdx.x] += red[0][threadIdx.x + s];
            red[1][threadIdx.x] += red[1][threadIdx.x + s];
            red[2][threadIdx.x] += red[2][threadIdx.x + s];
            red[3][threadIdx.x] += red[3][threadIdx.x + s];
        }
        __syncthreads();
    }
    if (threadIdx.x == 0) {
        sums[pairIdx * 4 + 0] = red[0][0];
        sums[pairIdx * 4 + 1] = red[1][0];
        sums[pairIdx * 4 + 2] = red[2][0];
        sums[pairIdx * 4 + 3] = red[3][0];
        const float* lg = logits + (size_t)(b * QQ + q) * CC;
        float l0 = lg[0], l1 = lg[1], l2 = lg[2];
        float mx  = fmaxf(l0, fmaxf(l1, l2));
        float s0 = __builtin_amdgcn_exp2f((l0 - mx) * LOG2E_F)
                 + __builtin_amdgcn_exp2f((l1 - mx) * LOG2E_F)
                 + __builtin_amdgcn_exp2f((l2 - mx) * LOG2E_F);
        float lse = mx + LN2_F * __builtin_amdgcn_logf(s0);
        int lab = labels[b * NN + t];
        float ll = (lab == 0) ? l0 : ((lab == 1) ? l1 : l2);
        cevals[pairIdx] = lse - ll;     // -log_softmax at label
    }
}

// ---------------------------------------------------------------------------
// Kernel D: fold 40 pair results into the 3 output scalars.
// ---------------------------------------------------------------------------
__global__ void final_kernel(const float* __restrict__ sums,
                             const float* __restrict__ cevals,
                             float* __restrict__ out)
{
    if (threadIdx.x != 0 || blockIdx.x != 0) return;
    float tot = 0.f, totce = 0.f, totdice = 0.f;
    for (int b = 0; b < BB; ++b) {
        float ceb = 0.f, diceacc = 0.f, bceacc = 0.f;
        for (int i = 0; i < NN; ++i) {
            int idx = b * NN + i;
            ceb += cevals[idx];
            float inter = sums[idx * 4 + 0];
            float psum  = sums[idx * 4 + 1];
            float tsum  = sums[idx * 4 + 2];
            diceacc += (2.f * inter + 1.f) / (psum + tsum + 1.f);
            bceacc  += sums[idx * 4 + 3];
        }
        ceb /= (float)NN;
        float diceb = 1.f - diceacc / (float)NN;
        float bceb  = bceacc / ((float)NN * (float)HWD);
        tot     += ceb + diceb + bceb;
        totce   += ceb;
        totdice += diceb;
    }
    out[0] = tot     / (float)BB;
    out[1] = totce   / (float)BB;
    out[2] = totdice / (float)BB;
}

// ---------------------------------------------------------------------------
extern "C" void kernel_launch(void* const* d_in, const int* in_sizes, int n_in,
                              void* d_out, int out_size, void* d_ws, size_t ws_size,
                              hipStream_t stream) {
    (void)in_sizes; (void)n_in; (void)out_size; (void)ws_size;
    const float* logits = (const float*)d_in[0];   // (B,Q,C)
    const float* pm     = (const float*)d_in[1];   // (B,Q,H,W)
    const int*   labels = (const int*)  d_in[2];   // (B,N)
    const float* tmk    = (const float*)d_in[3];   // (B,N,H,W)
    float* out = (float*)d_out;

    char*  ws     = (char*)d_ws;
    float* cost   = (float*)ws;                       // B*Q*N = 4000 f (16000 B)
    int*   qof    = (int*)(ws + 16000);               // 40 ints
    int*   tof    = (int*)(ws + 16160);               // 40 ints
    float* sums   = (float*)(ws + 16320);             // 160 f
    float* cevals = (float*)(ws + 16960);             // 40 f

    dim3 gridA((QQ + 15) / 16, BB);
    cost_kernel<<<gridA, 256, 0, stream>>>(logits, pm, labels, tmk, cost);
    hungarian_kernel<<<BB, 32, 0, stream>>>(cost, qof, tof);
    pair_kernel<<<BB * NN, 256, 0, stream>>>(logits, pm, labels, tmk, qof, tof, sums, cevals);
    final_kernel<<<1, 32, 0, stream>>>(sums, cevals, out);
}